// GNN_42356967473290
// MI455X (gfx1250) — compile-verified
//
#include <hip/hip_runtime.h>

typedef __attribute__((ext_vector_type(2))) float v2f;
typedef __attribute__((ext_vector_type(8))) float v8f;

#define NF 64

__device__ __forceinline__ void atomAddF(float* p, float v) {
  __hip_atomic_fetch_add(p, v, __ATOMIC_RELAXED, __HIP_MEMORY_SCOPE_AGENT);
}

// deg[i] = 1 (self loop)
__global__ void k_init_deg(float* __restrict__ deg, int n) {
  int i = blockIdx.x * blockDim.x + threadIdx.x;
  if (i < n) deg[i] = 1.0f;
}

__global__ void k_count_deg(const int* __restrict__ dst, float* __restrict__ deg, int E) {
  int e = blockIdx.x * blockDim.x + threadIdx.x;
  if (e < E) atomAddF(deg + dst[e], 1.0f);
}

// in place: deg -> d^{-1/2}; deg >= 1 always (self loops), so no zero guard needed
__global__ void k_dinv(float* __restrict__ deg, int n) {
  int i = blockIdx.x * blockDim.x + threadIdx.x;
  if (i < n) deg[i] = rsqrtf(deg[i]);
}

// C[M,64] = A[M,64] * W[64,64] via V_WMMA_F32_16X16X4_F32.
// One wave -> 16 rows x 64 cols (4 accumulators, A-frag reused 4x).
__global__ void k_gemm_wmma(const float* __restrict__ A, const float* __restrict__ W,
                            float* __restrict__ C, int M) {
  int wave = (int)((blockIdx.x * (unsigned)blockDim.x + threadIdx.x) >> 5);
  int lane = (int)(threadIdx.x & 31u);
  int m0 = wave * 16;
  if (m0 >= M) return;              // wave-uniform: EXEC stays all-ones for WMMA
  int half = lane >> 4;             // K pair select (0: K=0,1 ; 1: K=2,3)
  int lm   = lane & 15;             // M row (A) / N col (B)

  v8f acc0 = {}, acc1 = {}, acc2 = {}, acc3 = {};
  const float* arow = A + (size_t)(m0 + lm) * NF;

  for (int k0 = 0; k0 < NF; k0 += 4) {
    int ka = k0 + half * 2;
    v2f a;
    a.x = arow[ka];
    a.y = arow[ka + 1];
    const float* w0 = W + (size_t)ka * NF;  // row K=ka
    const float* w1 = w0 + NF;              // row K=ka+1
    v2f b0, b1, b2, b3;
    b0.x = w0[lm +  0]; b0.y = w1[lm +  0];
    b1.x = w0[lm + 16]; b1.y = w1[lm + 16];
    b2.x = w0[lm + 32]; b2.y = w1[lm + 32];
    b3.x = w0[lm + 48]; b3.y = w1[lm + 48];
    acc0 = __builtin_amdgcn_wmma_f32_16x16x4_f32(false, a, false, b0, (short)0, acc0, false, false);
    acc1 = __builtin_amdgcn_wmma_f32_16x16x4_f32(false, a, false, b1, (short)0, acc1, false, false);
    acc2 = __builtin_amdgcn_wmma_f32_16x16x4_f32(false, a, false, b2, (short)0, acc2, false, false);
    acc3 = __builtin_amdgcn_wmma_f32_16x16x4_f32(false, a, false, b3, (short)0, acc3, false, false);
  }

  // C/D layout: VGPR r, lanes 0-15 -> M=r, lanes 16-31 -> M=8+r, N=lm (+n0)
  float* crow = C + (size_t)(m0 + half * 8) * NF + lm;
#pragma unroll
  for (int r = 0; r < 8; ++r) {
    crow[(size_t)r * NF +  0] = acc0[r];
    crow[(size_t)r * NF + 16] = acc1[r];
    crow[(size_t)r * NF + 32] = acc2[r];
    crow[(size_t)r * NF + 48] = acc3[r];
  }
}

// hs = h * dinv[row] (in place); out = bias + hs * dinv[row]  (self-loop msg)
__global__ void k_scale_init(float* __restrict__ h, const float* __restrict__ dinv,
                             const float* __restrict__ bias, float* __restrict__ out, int n) {
  int idx = blockIdx.x * blockDim.x + threadIdx.x;   // (node, 4-feature group)
  if (idx >= n * 16) return;
  int node = idx >> 4;
  int g = (idx & 15) << 2;
  float di = dinv[node];
  float4 v = *(const float4*)(h + (size_t)node * NF + g);
  v.x *= di; v.y *= di; v.z *= di; v.w *= di;
  *(float4*)(h + (size_t)node * NF + g) = v;
  float4 o;
  o.x = bias[g + 0] + v.x * di;
  o.y = bias[g + 1] + v.y * di;
  o.z = bias[g + 2] + v.z * di;
  o.w = bias[g + 3] + v.w * di;
  *(float4*)(out + (size_t)node * NF + g) = o;
}

// out[dst] += hs[src] * dinv[dst]   (fp32 hardware atomics)
__global__ void k_scatter(const float* __restrict__ hs, const float* __restrict__ dinv,
                          const int* __restrict__ src, const int* __restrict__ dst,
                          float* __restrict__ out, int E) {
  long long idx = (long long)blockIdx.x * blockDim.x + threadIdx.x;
  if (idx >= (long long)E * 16) return;
  int e = (int)(idx >> 4);
  int g = ((int)idx & 15) << 2;
  int s = src[e];
  int d = dst[e];
  float nd = dinv[d];
  float4 v = *(const float4*)(hs + (size_t)s * NF + g);
  float* op = out + (size_t)d * NF + g;
  atomAddF(op + 0, v.x * nd);
  atomAddF(op + 1, v.y * nd);
  atomAddF(op + 2, v.z * nd);
  atomAddF(op + 3, v.w * nd);
}

__global__ void k_relu(float* __restrict__ x, int n4) {
  int i = blockIdx.x * blockDim.x + threadIdx.x;
  if (i >= n4) return;
  float4 v = ((const float4*)x)[i];
  v.x = fmaxf(v.x, 0.0f);
  v.y = fmaxf(v.y, 0.0f);
  v.z = fmaxf(v.z, 0.0f);
  v.w = fmaxf(v.w, 0.0f);
  ((float4*)x)[i] = v;
}

extern "C" void kernel_launch(void* const* d_in, const int* in_sizes, int n_in,
                              void* d_out, int out_size, void* d_ws, size_t ws_size,
                              hipStream_t stream) {
  const float* x  = (const float*)d_in[0];
  const int*   ei = (const int*)d_in[1];
  const float* W1 = (const float*)d_in[2];
  const float* b1 = (const float*)d_in[3];
  const float* W2 = (const float*)d_in[4];
  const float* b2 = (const float*)d_in[5];

  const int n = in_sizes[0] / NF;     // 100000 nodes (divisible by 16)
  const int E = in_sizes[1] / 2;      // 1.6M edges
  const int* src = ei;
  const int* dst = ei + E;

  float* out  = (float*)d_out;
  float* ws   = (float*)d_ws;
  float* dinv = ws;                                 // n floats
  float* h    = ws + n;                             // n*64 floats (hs in place)
  float* out1 = ws + n + (size_t)n * NF;            // n*64 floats

  const int B = 256;
  const int nB   = (n + B - 1) / B;
  const int eB   = (E + B - 1) / B;
  const int n16  = n * 16;
  const int n16B = (n16 + B - 1) / B;
  const long long sthreads = (long long)E * 16;
  const int sB = (int)((sthreads + B - 1) / B);
  const int waves = (n + 15) / 16;                  // one wave per 16-row strip
  const int gB = (waves + 7) / 8;                   // 8 waves per 256-thread block

  // degree -> d^{-1/2} (with self loops)
  k_init_deg<<<nB, B, 0, stream>>>(dinv, n);
  k_count_deg<<<eB, B, 0, stream>>>(dst, dinv, E);
  k_dinv<<<nB, B, 0, stream>>>(dinv, n);

  // layer 1
  k_gemm_wmma<<<gB, B, 0, stream>>>(x, W1, h, n);
  k_scale_init<<<n16B, B, 0, stream>>>(h, dinv, b1, out1, n);
  k_scatter<<<sB, B, 0, stream>>>(h, dinv, src, dst, out1, E);
  k_relu<<<n16B, B, 0, stream>>>(out1, n16);

  // layer 2
  k_gemm_wmma<<<gB, B, 0, stream>>>(out1, W2, h, n);
  k_scale_init<<<n16B, B, 0, stream>>>(h, dinv, b2, out, n);
  k_scatter<<<sB, B, 0, stream>>>(h, dinv, src, dst, out, E);
  k_relu<<<n16B, B, 0, stream>>>(out, n16);
}